// FFTCNNModule_27479200760451
// MI455X (gfx1250) — compile-verified
//
#include <hip/hip_runtime.h>

// CDNA5 (gfx1250) wave32 WMMA types
typedef __attribute__((ext_vector_type(2))) float v2f;   // f32 A/B operand (16x4 / 4x16)
typedef __attribute__((ext_vector_type(8))) float v8f;   // f32 C/D accumulator (16x16)

#define PI_F 3.14159265358979323846f

__device__ __forceinline__ v8f wmma4(v2f a, v2f b, v8f c) {
  // D = A(16x4) * B(4x16) + C ; f32 WMMA: no A/B neg (NEG=CNeg only), c_mod=0
  return __builtin_amdgcn_wmma_f32_16x16x4_f32(false, a, false, b, (short)0, c,
                                               false, false);
}

// One 1024-point complex FFT per wave32, single HBM pass.
// Four-step decomposition 1024 = 16 * 16 * 4:
//   phase1: Y[k1][n2]  = W1024^(k1*n2) * sum_n1 x[n1*64+n2] * W16^(n1*k1)
//   phase2: Z[k1][j1][m2] = W64^(j1*m2) * sum_m1 Y[k1][m1*4+m2] * W16^(m1*j1)
//   phase3: X[k1 + 16*j1 + 256*j2] = sum_m2 Z[k1][j1][m2] * W4^(m2*j2)
__global__ __launch_bounds__(256) void fft1024_wmma_kernel(
    const float* __restrict__ x, float* __restrict__ out, int nrows) {
  __shared__ float lds[8 * 2048];            // 8 waves * (1024 re + 1024 im)
  const int lane = threadIdx.x & 31;
  const int wave = threadIdx.x >> 5;
  const int row  = blockIdx.x * 8 + wave;
  if (row >= nrows) return;                  // wave-uniform: EXEC all-ones inside

  float* lre = &lds[wave * 2048];
  float* lim = lre + 1024;

  const int mn   = lane & 15;                // A-matrix row M == B/C/D column N
  const int half = lane >> 4;                // lane half selects K/M sub-block

  // ---- constant A operands: F16[m][k] = exp(-2*pi*i*m*k/16) ----
  // A layout (16x4 f32): lane m = lane&15, K = vgpr + 2*half; chunk i -> K = 4i..4i+3
  v2f Ar[4], Ai[4], AiN[4];
#pragma unroll
  for (int i = 0; i < 4; ++i) {
    const int k0 = 4 * i + 2 * half;
    float s0, c0, s1, c1;
    __sincosf(-(PI_F / 8.0f) * (float)(mn * k0), &s0, &c0);
    __sincosf(-(PI_F / 8.0f) * (float)(mn * (k0 + 1)), &s1, &c1);
    v2f vr = {c0, c1}; Ar[i] = vr;
    v2f vi = {s0, s1}; Ai[i] = vi;
    v2f vn = {-s0, -s1}; AiN[i] = vn;
  }

  // ---- stage one row into LDS (split re/im), coalesced 128-bit loads ----
  {
    const float4* src = (const float4*)(x + (size_t)row * 2048);
#pragma unroll
    for (int i = 0; i < 16; ++i) {
      const int j = i * 32 + lane;           // complex pair index (2 per float4)
      const float4 v = src[j];
      lre[2 * j]     = v.x;  lim[2 * j]     = v.y;
      lre[2 * j + 1] = v.z;  lim[2 * j + 1] = v.w;
    }
  }

  // ---- phase 1: 64 DFT-16s (stride 64), in-place per 16-column tile ----
#pragma unroll
  for (int t = 0; t < 4; ++t) {
    const int n2 = 16 * t + mn;
    v8f aR = {0,0,0,0,0,0,0,0}, aI = {0,0,0,0,0,0,0,0};
#pragma unroll
    for (int i = 0; i < 4; ++i) {
      // B layout (4x16 f32): K = vgpr + 2*half, N = lane&15
      const int r0 = (4 * i + 2 * half) * 64 + n2;
      v2f bre = { lre[r0], lre[r0 + 64] };
      v2f bim = { lim[r0], lim[r0 + 64] };
      aR = wmma4(Ar[i],  bre, aR);           // Yr += Fr*Xr
      aR = wmma4(AiN[i], bim, aR);           // Yr -= Fi*Xi
      aI = wmma4(Ar[i],  bim, aI);           // Yi += Fr*Xi
      aI = wmma4(Ai[i],  bre, aI);           // Yi += Fi*Xr
    }
#pragma unroll
    for (int v = 0; v < 8; ++v) {            // C/D layout: M = vgpr + 8*half
      const int k1 = v + 8 * half;
      float s, c;
      __sincosf(-(2.0f * PI_F / 1024.0f) * (float)(k1 * n2), &s, &c);
      const float yr = aR[v], yi = aI[v];
      lre[k1 * 64 + n2] = yr * c - yi * s;
      lim[k1 * 64 + n2] = yr * s + yi * c;
    }
  }

  // ---- phase 2: 64 DFT-16s (stride 4 within each k1 row) ----
#pragma unroll
  for (int t = 0; t < 4; ++t) {
    const int q  = 16 * t + mn;              // instance -> (k1, m2)
    const int k1 = q >> 2;
    const int m2 = q & 3;
    const int base = k1 * 64 + m2;
    v8f aR = {0,0,0,0,0,0,0,0}, aI = {0,0,0,0,0,0,0,0};
#pragma unroll
    for (int i = 0; i < 4; ++i) {
      const int r0 = base + (4 * i + 2 * half) * 4;
      v2f bre = { lre[r0], lre[r0 + 4] };
      v2f bim = { lim[r0], lim[r0 + 4] };
      aR = wmma4(Ar[i],  bre, aR);
      aR = wmma4(AiN[i], bim, aR);
      aI = wmma4(Ar[i],  bim, aI);
      aI = wmma4(Ai[i],  bre, aI);
    }
#pragma unroll
    for (int v = 0; v < 8; ++v) {
      const int j1 = v + 8 * half;
      float s, c;
      __sincosf(-(2.0f * PI_F / 64.0f) * (float)(j1 * m2), &s, &c);
      const float yr = aR[v], yi = aI[v];
      // XOR swizzle on j1 (bijection within the in-place-safe column set)
      // makes phase-3 ds_load_b128 bank-conflict-free.
      const int dst = k1 * 64 + ((j1 ^ k1) * 4) + m2;
      lre[dst] = yr * c - yi * s;
      lim[dst] = yr * s + yi * c;
    }
  }

  // ---- phase 3: 256 radix-4 butterflies + coalesced stores ----
  float2* o = (float2*)out + (size_t)row * 1024;
#pragma unroll
  for (int r = 0; r < 8; ++r) {
    const int p  = r * 32 + lane;            // p = k1 + 16*j1
    const int k1 = p & 15;
    const int j1 = p >> 4;
    const int b0 = k1 * 64 + ((j1 ^ k1) * 4);
    const float4 ar = *(const float4*)&lre[b0];   // z.re, m2 = 0..3
    const float4 br = *(const float4*)&lim[b0];   // z.im, m2 = 0..3
    // W4 = -i:  X[j2] = sum z[m2] * (-i)^(m2*j2)
    float2 X0 = { ar.x + ar.y + ar.z + ar.w,  br.x + br.y + br.z + br.w };
    float2 X1 = { ar.x + br.y - ar.z - br.w,  br.x - ar.y - br.z + ar.w };
    float2 X2 = { ar.x - ar.y + ar.z - ar.w,  br.x - br.y + br.z - br.w };
    float2 X3 = { ar.x - br.y - ar.z + br.w,  br.x + ar.y - br.z - ar.w };
    o[p]       = X0;                          // out index = p + 256*j2
    o[p + 256] = X1;
    o[p + 512] = X2;
    o[p + 768] = X3;
  }
}

extern "C" void kernel_launch(void* const* d_in, const int* in_sizes, int n_in,
                              void* d_out, int out_size, void* d_ws, size_t ws_size,
                              hipStream_t stream) {
  (void)n_in; (void)out_size; (void)d_ws; (void)ws_size;
  const float* x = (const float*)d_in[0];
  float* out = (float*)d_out;
  const int nrows  = in_sizes[0] / 2048;     // B rows of 1024 complex
  const int blocks = (nrows + 7) / 8;        // 8 waves (rows) per 256-thread block
  fft1024_wmma_kernel<<<blocks, 256, 0, stream>>>(x, out, nrows);
}